// EdgeFeatureGNN_35923106463755
// MI455X (gfx1250) — compile-verified
//
#include <hip/hip_runtime.h>

#define N_NODES 50000
#define N_EDGES 400000
// NODE_IN = 16, EDGE_IN = 8, HID = 16

typedef float v2f __attribute__((ext_vector_type(2)));
typedef float v8f __attribute__((ext_vector_type(8)));
typedef unsigned int u32x4 __attribute__((ext_vector_type(4)));
typedef int i32x4 __attribute__((ext_vector_type(4)));
typedef int i32x8 __attribute__((ext_vector_type(8)));

__device__ __forceinline__ v8f wmma_f32_4(v2f a, v2f b, v8f c) {
    // V_WMMA_F32_16X16X4_F32 : D = A(16x4) * B(4x16) + C(16x16)
    return __builtin_amdgcn_wmma_f32_16x16x4_f32(false, a, false, b, (short)0, c,
                                                 false, false);
}

__device__ __forceinline__ void atomic_add_f32(float* p, float v) {
    unsafeAtomicAdd(p, v);  // global_atomic_add_f32
}

// ---- Tensor Data Mover: 1-D dword tile, optional LDS padding --------------
// D# per CDNA5 ISA ch.8: group0 {count=1, lds_addr, global_addr, type=2},
// group1 {data_size=4B, pad cfg, tensor_dim0=n, tile_dim0=n, stride0=n}.
// pad_interval code 2 = every 8 dwords, pad_amount code 0 = +1 dword
// (gives LDS row stride 9 for the 8-float edge_attr rows).
// This toolchain exposes the 6-arg builtin (clang-23 / therock headers).
__device__ __forceinline__ void tdm_load_1d(unsigned int lds_addr,
                                            const void* gptr,
                                            unsigned int n_dwords,
                                            bool pad, unsigned int pad_interval,
                                            unsigned int pad_amount) {
    unsigned long long ga = (unsigned long long)gptr;
    u32x4 g0;
    g0.x = 1u;                                   // count=1, no gather, user mode
    g0.y = lds_addr;                             // LDS byte address
    g0.z = (unsigned int)ga;                     // global_addr[31:0]
    g0.w = (unsigned int)(ga >> 32) | (2u << 30);// global_addr[56:32] | type=2
    unsigned int w0 = (2u << 16);                // data_size=2 (4 bytes), mask=0
    if (pad) w0 |= (1u << 20) | (pad_interval << 22) | (pad_amount << 25);
    i32x8 g1;
    g1[0] = (int)w0;
    g1[1] = (int)((n_dwords & 0xFFFFu) << 16);   // tensor_dim0[15:0]
    g1[2] = (int)((n_dwords >> 16) | (1u << 16));// tensor_dim0[31:16] | tensor_dim1=1
    g1[3] = (int)((n_dwords & 0xFFFFu) << 16);   // tile_dim0 (<=65535)
    g1[4] = 0;                                   // tile_dim1=0, tile_dim2=0
    g1[5] = (int)n_dwords;                       // tensor_dim0_stride[31:0]
    g1[6] = 0;
    g1[7] = 0;
    i32x4 gz4 = {0, 0, 0, 0};
    i32x8 gz8 = {0, 0, 0, 0, 0, 0, 0, 0};
    __builtin_amdgcn_tensor_load_to_lds(g0, g1, gz4, gz4, gz8, 0);
}

// ---------------------------------------------------------------- utilities
__global__ void zero_kernel(float4* __restrict__ p, int n4) {
    int i = blockIdx.x * blockDim.x + threadIdx.x;
    if (i < n4) p[i] = make_float4(0.f, 0.f, 0.f, 0.f);
}

__global__ void count_kernel(const long long* __restrict__ ei, float* __restrict__ cnt) {
    int e = blockIdx.x * blockDim.x + threadIdx.x;
    if (e < N_EDGES) {
        int d = (int)ei[N_EDGES + e];
        atomic_add_f32(&cnt[d], 1.0f);
    }
}

// ------------------------------------------------ edge message kernel (WMMA)
// One wave = 16 edges. Block = 256 threads = 8 waves = 128 edges.
//   h   = relu(ea@w1+b1)                       (2x wmma, K=8)
//   C_t = h @ w2[:,16t:16t+16]                 (4x wmma each, 16 tiles)
//   msg[e,:] += x_src[e,t]*(C_t[e,:]+b2[t,:])  (bmm collapses into FMA)
// then scatter-adds msg into msum[dst].
__launch_bounds__(256)
__global__ void edge_msg_kernel(const float* __restrict__ xin,
                                const float* __restrict__ edge_attr,
                                const long long* __restrict__ ei,
                                const float* __restrict__ w1,
                                const float* __restrict__ b1,
                                const float* __restrict__ w2,
                                const float* __restrict__ b2,
                                float* __restrict__ msum) {
    __shared__ float s_w1[8 * 16];
    __shared__ float s_b1[16];
    __shared__ float s_w2[16 * 256];     // 16 KB, staged by TDM
    __shared__ float s_b2[256];
    __shared__ float s_ea[8][16][9];     // stride 9, TDM pad feature fills it
    __shared__ float s_h [8][16][17];    // stride 17 (conflict-free)
    __shared__ float s_xs[8][16][17];
    __shared__ int   s_src[128];
    __shared__ int   s_dst[128];

    const int t  = threadIdx.x;
    const int eb = blockIdx.x * 128;     // E % 128 == 0, no tail

    // ---- TDM DMA: w2 (16KB contiguous) + edge_attr tile (4KB, pad->stride 9)
    if (t < 32) {                        // wave-uniform; TDM issued once
        tdm_load_1d((unsigned int)(uintptr_t)s_w2, w2, 16 * 256, false, 0, 0);
        tdm_load_1d((unsigned int)(uintptr_t)&s_ea[0][0][0],
                    edge_attr + (size_t)eb * 8, 128 * 8, true, 2, 0);
        __builtin_amdgcn_s_wait_tensorcnt(0);
    }
    // ---- small staging in parallel with the DMAs ----
    s_b2[t] = b2[t];
    if (t < 128) s_w1[t] = w1[t];
    if (t < 16)  s_b1[t] = b1[t];
    if (t < 128) {
        s_src[t] = (int)ei[eb + t];
        s_dst[t] = (int)ei[N_EDGES + eb + t];
    }
    __syncthreads();

    // ---- gather source node features: four b128 requests per 64B row ----
    for (int i = t; i < 512; i += 256) {
        int el = i >> 2, q = i & 3;
        float4 v = ((const float4*)(xin + (size_t)s_src[el] * 16))[q];
        float* d = &s_xs[el >> 4][el & 15][q * 4];
        d[0] = v.x; d[1] = v.y; d[2] = v.z; d[3] = v.w;
    }
    __syncthreads();

    const int wv   = t >> 5;
    const int lane = t & 31;
    const int m    = lane & 15;   // A-matrix row  (edge within tile)
    const int hi   = lane >> 4;   // half-wave select
    const int n    = lane & 15;   // B/C/D column

    // ---- h = relu(EA @ w1 + b1), K = 8 ----
    v8f ch = {};
#pragma unroll
    for (int s = 0; s < 2; ++s) {
        v2f a, b;
        a.x = s_ea[wv][m][4 * s + 2 * hi + 0];
        a.y = s_ea[wv][m][4 * s + 2 * hi + 1];
        b.x = s_w1[(4 * s + 2 * hi + 0) * 16 + n];
        b.y = s_w1[(4 * s + 2 * hi + 1) * 16 + n];
        ch = wmma_f32_4(a, b, ch);
    }
    {
        float b1v = s_b1[n];
#pragma unroll
        for (int r = 0; r < 8; ++r)
            s_h[wv][r + 8 * hi][n] = fmaxf(ch[r] + b1v, 0.0f);
    }
    __syncthreads();

    // ---- preload A fragments of h (K = 16 -> 4 steps) ----
    v2f ah[4];
#pragma unroll
    for (int s = 0; s < 4; ++s) {
        ah[s].x = s_h[wv][m][4 * s + 2 * hi + 0];
        ah[s].y = s_h[wv][m][4 * s + 2 * hi + 1];
    }

    // ---- 16 column tiles of h@w2; fold x_src multiply into accumulation ----
    float macc[8] = {0, 0, 0, 0, 0, 0, 0, 0};
#pragma unroll
    for (int t16 = 0; t16 < 16; ++t16) {
        v8f cc = {};
#pragma unroll
        for (int s = 0; s < 4; ++s) {
            v2f b;
            b.x = s_w2[(4 * s + 2 * hi + 0) * 256 + t16 * 16 + n];
            b.y = s_w2[(4 * s + 2 * hi + 1) * 256 + t16 * 16 + n];
            cc = wmma_f32_4(ah[s], b, cc);
        }
        float b2v = s_b2[t16 * 16 + n];
#pragma unroll
        for (int r = 0; r < 8; ++r)
            macc[r] += s_xs[wv][r + 8 * hi][t16] * (cc[r] + b2v);
    }

    // ---- scatter-add messages into destination nodes ----
#pragma unroll
    for (int r = 0; r < 8; ++r) {
        int el = wv * 16 + r + 8 * hi;
        int d  = s_dst[el];
        atomic_add_f32(&msum[(size_t)d * 16 + n], macc[r]);
    }
}

// ------------------------------------------------ node update kernel (WMMA)
// hout = relu(msum/max(cnt,1) + xin@root + bias). Runs in-place
// (hout == msum): each thread reads & writes the same element.
__launch_bounds__(256)
__global__ void node_update_kernel(const float* __restrict__ xin,
                                   const float* __restrict__ msum,
                                   const float* __restrict__ cnt,
                                   const float* __restrict__ root,
                                   const float* __restrict__ bias,
                                   float* __restrict__ hout) {
    __shared__ float s_root[16 * 16];
    __shared__ float s_bias[16];
    __shared__ float s_x[8][16][17];

    const int t  = threadIdx.x;
    const int nb = blockIdx.x * 128;

    s_root[t] = root[t];
    if (t < 16) s_bias[t] = bias[t];
    for (int i = t; i < 512; i += 256) {
        int nl = i >> 2, q = i & 3;
        int node = nb + nl;
        float4 v = make_float4(0.f, 0.f, 0.f, 0.f);
        if (node < N_NODES) v = ((const float4*)(xin + (size_t)node * 16))[q];
        float* d = &s_x[nl >> 4][nl & 15][q * 4];
        d[0] = v.x; d[1] = v.y; d[2] = v.z; d[3] = v.w;
    }
    __syncthreads();

    const int wv   = t >> 5;
    const int lane = t & 31;
    const int m    = lane & 15;
    const int hi   = lane >> 4;
    const int n    = lane & 15;
    const int tn   = nb + wv * 16;
    if (tn >= N_NODES) return;  // wave-uniform: EXEC stays all-ones below

    v8f c = {};
#pragma unroll
    for (int s = 0; s < 4; ++s) {
        v2f a, b;
        a.x = s_x[wv][m][4 * s + 2 * hi + 0];
        a.y = s_x[wv][m][4 * s + 2 * hi + 1];
        b.x = s_root[(4 * s + 2 * hi + 0) * 16 + n];
        b.y = s_root[(4 * s + 2 * hi + 1) * 16 + n];
        c = wmma_f32_4(a, b, c);
    }

    float bv = s_bias[n];
#pragma unroll
    for (int r = 0; r < 8; ++r) {
        int node = tn + r + 8 * hi;
        if (node < N_NODES) {
            float inv = 1.0f / fmaxf(cnt[node], 1.0f);
            float val = c[r] + bv + msum[(size_t)node * 16 + n] * inv;
            hout[(size_t)node * 16 + n] = fmaxf(val, 0.0f);
        }
    }
}

// ------------------------------------------------ q head
__global__ void q_kernel(const float* __restrict__ h, const float* __restrict__ qw,
                         const float* __restrict__ qb, float* __restrict__ out) {
    int i = blockIdx.x * blockDim.x + threadIdx.x;
    if (i < N_NODES) {
        float acc = qb[0];
        const float4* hr = (const float4*)(h + (size_t)i * 16);
#pragma unroll
        for (int q4 = 0; q4 < 4; ++q4) {
            float4 v = hr[q4];
            acc += v.x * qw[q4 * 4 + 0] + v.y * qw[q4 * 4 + 1] +
                   v.z * qw[q4 * 4 + 2] + v.w * qw[q4 * 4 + 3];
        }
        out[i] = acc;
    }
}

// ---------------------------------------------------------------- launcher
extern "C" void kernel_launch(void* const* d_in, const int* in_sizes, int n_in,
                              void* d_out, int out_size, void* d_ws, size_t ws_size,
                              hipStream_t stream) {
    (void)in_sizes; (void)n_in; (void)out_size; (void)ws_size;

    const float* x     = (const float*)d_in[0];
    const float* ea    = (const float*)d_in[1];
    const float* e1_w1 = (const float*)d_in[2];
    const float* e1_b1 = (const float*)d_in[3];
    const float* e1_w2 = (const float*)d_in[4];
    const float* e1_b2 = (const float*)d_in[5];
    const float* root1 = (const float*)d_in[6];
    const float* bias1 = (const float*)d_in[7];
    const float* e2_w1 = (const float*)d_in[8];
    const float* e2_b1 = (const float*)d_in[9];
    const float* e2_w2 = (const float*)d_in[10];
    const float* e2_b2 = (const float*)d_in[11];
    const float* root2 = (const float*)d_in[12];
    const float* bias2 = (const float*)d_in[13];
    const float* q_w   = (const float*)d_in[14];
    const float* q_b   = (const float*)d_in[15];
    const long long* eidx = (const long long*)d_in[16];

    // workspace: [cnt N][sum1 16N][sum2 16N]; h1/h2 computed in-place over sums
    float* ws   = (float*)d_ws;
    float* cnt  = ws;
    float* sum1 = cnt + N_NODES;
    float* sum2 = sum1 + 16 * N_NODES;
    float* h1   = sum1;  // in-place node update
    float* h2   = sum2;

    const int ZN4 = N_NODES * 33 / 4;  // 33N divisible by 4
    zero_kernel<<<(ZN4 + 255) / 256, 256, 0, stream>>>((float4*)ws, ZN4);
    count_kernel<<<(N_EDGES + 255) / 256, 256, 0, stream>>>(eidx, cnt);

    edge_msg_kernel<<<N_EDGES / 128, 256, 0, stream>>>(x, ea, eidx,
                                                       e1_w1, e1_b1, e1_w2, e1_b2, sum1);
    node_update_kernel<<<(N_NODES + 127) / 128, 256, 0, stream>>>(x, sum1, cnt,
                                                                  root1, bias1, h1);
    edge_msg_kernel<<<N_EDGES / 128, 256, 0, stream>>>(h1, ea, eidx,
                                                       e2_w1, e2_b1, e2_w2, e2_b2, sum2);
    node_update_kernel<<<(N_NODES + 127) / 128, 256, 0, stream>>>(h1, sum2, cnt,
                                                                  root2, bias2, h2);
    q_kernel<<<(N_NODES + 255) / 256, 256, 0, stream>>>(h2, q_w, q_b, (float*)d_out);
}